// dMaSIFSiteEmbed_39298950758858
// MI455X (gfx1250) — compile-verified
//
#include <hip/hip_runtime.h>
#include <math.h>

typedef __attribute__((ext_vector_type(2))) float v2f;
typedef __attribute__((ext_vector_type(8))) float v8f;

#define NPTS       4096
#define INV_R      (1.0f/9.0f)
#define CONV_SCALE 0.07856742013183862f   /* 1/(sqrt(2)*9) */

__device__ __forceinline__ float leaky(float x){ return x >= 0.f ? x : 0.2f*x; }

/* ---------------- K0: ordering scores: leaky(f@W1^T+b1)@W2^T+b2 ---------------- */
__global__ __launch_bounds__(256) void k_scores(
    const float* __restrict__ feats, const float* __restrict__ w1,
    const float* __restrict__ b1, const float* __restrict__ w2,
    const float* __restrict__ b2, float* __restrict__ scores)
{
  __shared__ float sw1[256], sb1[16], sw2[16];
  int tid = threadIdx.x;
  sw1[tid] = w1[tid];
  if (tid < 16){ sb1[tid]=b1[tid]; sw2[tid]=w2[tid]; }
  __syncthreads();
  int i = blockIdx.x*256 + tid;
  float fv[16];
  #pragma unroll
  for (int k=0;k<16;++k) fv[k]=feats[i*16+k];
  float s = b2[0];
  #pragma unroll
  for (int u=0;u<16;++u){
    float t = sb1[u];
    #pragma unroll
    for (int k=0;k<16;++k) t = fmaf(fv[k], sw1[u*16+k], t);
    s = fmaf(leaky(t), sw2[u], s);
  }
  scores[i] = s;
}

/* ---------------- K1: load_mesh -> nuv frames (O(N^2) windowed reduction) ------ */
__global__ __launch_bounds__(256) void k_nuv(
    const float* __restrict__ xyz, const float* __restrict__ nrm,
    const float* __restrict__ scores, float* __restrict__ nuv)
{
  __shared__ float sp[256][3], sn[256][3], ss[256];
  int tid = threadIdx.x;
  int i = blockIdx.x*256 + tid;
  float pi[3], ni[3];
  #pragma unroll
  for (int d=0;d<3;++d){ pi[d]=xyz[i*3+d]*INV_R; ni[d]=nrm[i*3+d]; }
  float s = ni[2] >= 0.f ? 1.f : -1.f;
  float a = -1.f/(s+ni[2]);
  float b = ni[0]*ni[1]*a;
  float u0 = 1.f + s*ni[0]*ni[0]*a, u1 = s*b,                u2 = -s*ni[0];
  float v0 = b,                     v1 = s + ni[1]*ni[1]*a,  v2 = -ni[1];
  float ov0=0.f, ov1=0.f;
  for (int jt=0; jt<NPTS/256; ++jt){
    __syncthreads();
    int j = jt*256+tid;
    #pragma unroll
    for (int d=0;d<3;++d){ sp[tid][d]=xyz[j*3+d]*INV_R; sn[tid][d]=nrm[j*3+d]; }
    ss[tid]=scores[j];
    __syncthreads();
    for (int jj=0; jj<256; ++jj){
      float d0=sp[jj][0]-pi[0], d1=sp[jj][1]-pi[1], d2v=sp[jj][2]-pi[2];
      float dd = d0*d0+d1*d1+d2v*d2v;
      float cs = ni[0]*sn[jj][0]+ni[1]*sn[jj][1]+ni[2]*sn[jj][2];
      float t = 2.f - cs;
      float w = __expf(-dd*t*t) * ss[jj];
      ov0 = fmaf(w, u0*d0+u1*d1+u2*d2v, ov0);
      ov1 = fmaf(w, v0*d0+v1*d1+v2*d2v, ov1);
    }
  }
  ov0 += 1e-5f; ov1 += 1e-5f;
  float nv = fmaxf(sqrtf(ov0*ov0+ov1*ov1), 1e-12f);
  float ex = ov0/nv, ey = ov1/nv;
  float* o = nuv + (size_t)i*9;
  o[0]=ni[0];          o[1]=ni[1];          o[2]=ni[2];
  o[3]=ex*u0+ey*v0;    o[4]=ex*u1+ey*v1;    o[5]=ex*u2+ey*v2;
  o[6]=-ey*u0+ex*v0;   o[7]=-ey*u1+ex*v1;   o[8]=-ey*u2+ex*v2;
}

/* ---------------- K2: input MLP + GroupNorm (global stats -> single WG) -------- */
__global__ __launch_bounds__(1024) void k_inmlp_gn(
    const float* __restrict__ feats,
    const float* __restrict__ w1, const float* __restrict__ b1,
    const float* __restrict__ w2, const float* __restrict__ b2,
    const float* __restrict__ gam, const float* __restrict__ bet,
    float* __restrict__ hout)
{
  __shared__ float sw1[256], sw2[256], sb1[16], sb2[16], sg[16], sbt[16];
  __shared__ float gsum[4], gss[4], mean[4], rstd[4];
  int tid = threadIdx.x;
  if (tid<256){ sw1[tid]=w1[tid]; sw2[tid]=w2[tid]; }
  if (tid<16){ sb1[tid]=b1[tid]; sb2[tid]=b2[tid]; sg[tid]=gam[tid]; sbt[tid]=bet[tid]; }
  if (tid<4){ gsum[tid]=0.f; gss[tid]=0.f; }
  __syncthreads();
  float t2s[4][16];
  float ls[4]={0,0,0,0}, lq[4]={0,0,0,0};
  for (int k=0;k<4;++k){
    int pid = k*1024 + tid;
    float fv[16];
    #pragma unroll
    for (int c=0;c<16;++c) fv[c]=feats[pid*16+c];
    float t1[16];
    #pragma unroll
    for (int u=0;u<16;++u){
      float t = sb1[u];
      #pragma unroll
      for (int c=0;c<16;++c) t = fmaf(fv[c], sw1[u*16+c], t);
      t1[u] = leaky(t);
    }
    #pragma unroll
    for (int c=0;c<16;++c){
      float t = sb2[c];
      #pragma unroll
      for (int u=0;u<16;++u) t = fmaf(t1[u], sw2[c*16+u], t);
      t = leaky(t);
      t2s[k][c] = t;
      ls[c>>2] += t; lq[c>>2] = fmaf(t,t,lq[c>>2]);
    }
  }
  #pragma unroll
  for (int g=0; g<4; ++g){ atomicAdd(&gsum[g], ls[g]); atomicAdd(&gss[g], lq[g]); }
  __syncthreads();
  if (tid<4){
    float m = gsum[tid] * (1.0f/16384.0f);
    float v = gss[tid] * (1.0f/16384.0f) - m*m;
    mean[tid]=m; rstd[tid]=rsqrtf(v + 1e-5f);
  }
  __syncthreads();
  for (int k=0;k<4;++k){
    int pid = k*1024+tid;
    #pragma unroll
    for (int c=0;c<16;++c){
      int g = c>>2;
      hout[pid*16+c] = fmaf((t2s[k][c]-mean[g])*rstd[g], sg[c], sbt[c]);
    }
  }
}

/* ---------------- K3: main dMaSIF conv — WMMA f32 16x16x4 (K=8 via 2) ----------
 * One WG per 16 queries.  Per 16-neighbor tile:
 *   pair phase : 256 threads, one (b,n) pair each -> window, g[8] into LDS
 *   WMMA phase : 8 waves x 2 row-tiles; chunk o: D = g(16x8) @ cv_a2^T(8x16),
 *                then acc += relu(D+b2) * (window(row)*h(row,col))   (co-exec VALU)
 * B tiles + cv_b2 columns live in registers for the whole kernel.
 */
__global__ __launch_bounds__(256) void k_conv(
    const float* __restrict__ xyz, const float* __restrict__ nrm,
    const float* __restrict__ nuv, const float* __restrict__ hmat,
    const float* __restrict__ cva1, const float* __restrict__ cvb1,
    const float* __restrict__ cva2, const float* __restrict__ cvb2,
    float* __restrict__ y)
{
  __shared__ float qp[16][3];
  __shared__ float qnuv[16][9];
  __shared__ float wnd[16][17];
  __shared__ float gls[256][8];
  const int tid  = threadIdx.x;
  const int lane = tid & 31;
  const int wave = tid >> 5;
  const int col  = lane & 15;
  const int hi   = lane >> 4;
  const int btile = blockIdx.x;

  /* Persistent B operands: chunk o covers output columns f = o*16 + h.
   * WMMA#1 uses K=c0..3  -> (B0,B1); WMMA#2 uses K=c4..7 -> (B2,B3).
   * B layout (4x16, 32-bit): vreg0 = {K0 | K2}, vreg1 = {K1 | K3} across lane halves. */
  float B0[16], B1[16], B2[16], B3[16], breg[16];
  #pragma unroll
  for (int o=0;o<16;++o){
    int f = o*16 + col;
    B0[o] = cva2[f*8 + (hi ? 2 : 0)];
    B1[o] = cva2[f*8 + (hi ? 3 : 1)];
    B2[o] = cva2[f*8 + (hi ? 6 : 4)];
    B3[o] = cva2[f*8 + (hi ? 7 : 5)];
    breg[o] = cvb2[f];
  }
  /* tiny conv weights: uniform addresses -> scalar loads/SGPRs */
  float a1[8][3], b1v[8];
  #pragma unroll
  for (int c=0;c<8;++c){
    a1[c][0]=cva1[c*3+0]; a1[c][1]=cva1[c*3+1]; a1[c][2]=cva1[c*3+2];
    b1v[c]=cvb1[c];
  }
  if (tid < 16){
    int qi = btile*16 + tid;
    #pragma unroll
    for (int d=0; d<3; ++d) qp[tid][d] = xyz[qi*3+d]*CONV_SCALE;
    #pragma unroll
    for (int k=0;k<9;++k) qnuv[tid][k] = nuv[qi*9+k];
  }
  float acc[2][16];
  #pragma unroll
  for (int r=0;r<2;++r)
    #pragma unroll
    for (int o=0;o<16;++o) acc[r][o]=0.f;

  const int bl = tid >> 4, nl = tid & 15;
  __syncthreads();

  for (int nt=0; nt<NPTS/16; ++nt){
    const int n0 = nt*16;
    /* ---- pair phase ---- */
    int j = n0 + nl;
    float pj0 = xyz[j*3+0]*CONV_SCALE, pj1 = xyz[j*3+1]*CONV_SCALE, pj2 = xyz[j*3+2]*CONV_SCALE;
    float nj0 = nrm[j*3+0], nj1 = nrm[j*3+1], nj2 = nrm[j*3+2];
    float d0 = pj0 - qp[bl][0], d1 = pj1 - qp[bl][1], d2v = pj2 - qp[bl][2];
    float dd = d0*d0 + d1*d1 + d2v*d2v;
    float cs = qnuv[bl][0]*nj0 + qnuv[bl][1]*nj1 + qnuv[bl][2]*nj2;
    float t  = 2.f - cs;
    float w  = __expf(-dd*t*t);
    float X0 = qnuv[bl][0]*d0 + qnuv[bl][1]*d1 + qnuv[bl][2]*d2v;
    float X1 = qnuv[bl][3]*d0 + qnuv[bl][4]*d1 + qnuv[bl][5]*d2v;
    float X2 = qnuv[bl][6]*d0 + qnuv[bl][7]*d1 + qnuv[bl][8]*d2v;
    float g[8];
    #pragma unroll
    for (int c=0;c<8;++c)
      g[c] = fmaxf(fmaf(a1[c][0],X0, fmaf(a1[c][1],X1, fmaf(a1[c][2],X2, b1v[c]))), 0.f);
    __syncthreads();                 /* previous WMMA phase done with wnd/gls */
    wnd[bl][nl] = w;
    #pragma unroll
    for (int c=0;c<8;++c) gls[tid][c] = g[c];
    if (nt+1 < NPTS/16) __builtin_prefetch(&hmat[(n0+16)*16], 0, 1);
    __syncthreads();
    /* ---- WMMA phase ---- */
    float hcol[8];
    #pragma unroll
    for (int v=0;v<8;++v) hcol[v] = hmat[(n0 + v + 8*hi)*16 + col];
    #pragma unroll
    for (int ri=0; ri<2; ++ri){
      const int r = wave + 8*ri;
      float W[8];
      #pragma unroll
      for (int v=0;v<8;++v) W[v] = wnd[r][v + 8*hi] * hcol[v];
      const v2f* gp = (const v2f*)(&gls[r*16 + col][0]);
      v2f alo = gp[hi];        /* rows=n_local: {c0,c1} | {c2,c3} */
      v2f ahi = gp[2 + hi];    /* {c4,c5} | {c6,c7} */
      #pragma unroll
      for (int o=0;o<16;++o){
        v2f blo; blo[0]=B0[o]; blo[1]=B1[o];
        v2f bhi; bhi[0]=B2[o]; bhi[1]=B3[o];
        v8f dmat = {0.f,0.f,0.f,0.f,0.f,0.f,0.f,0.f};
        dmat = __builtin_amdgcn_wmma_f32_16x16x4_f32(false, alo, false, blo, (short)0, dmat, false, false);
        dmat = __builtin_amdgcn_wmma_f32_16x16x4_f32(false, ahi, false, bhi, (short)0, dmat, false, false);
        float av = acc[ri][o];
        #pragma unroll
        for (int v=0;v<8;++v)
          av = fmaf(fmaxf(dmat[v] + breg[o], 0.f), W[v], av);
        acc[ri][o] = av;
      }
    }
  }
  /* final cross-lane totals; plain stores -> deterministic */
  #pragma unroll
  for (int ri=0; ri<2; ++ri){
    #pragma unroll
    for (int o=0;o<16;++o){
      float v = acc[ri][o];
      v += __shfl_xor(v, 16);
      v += __shfl_xor(v, 8);
      v += __shfl_xor(v, 4);
      v += __shfl_xor(v, 2);
      v += __shfl_xor(v, 1);
      if (lane == 0) y[(btile*16 + wave + 8*ri)*16 + o] = v;
    }
  }
}

/* ---------------- K4: output MLP + GroupNorm + ll layers + lt skip ------------- */
__global__ __launch_bounds__(1024) void k_post(
    const float* __restrict__ y, const float* __restrict__ feats,
    const float* __restrict__ nw1, const float* __restrict__ nb1,
    const float* __restrict__ nw2, const float* __restrict__ nb2,
    const float* __restrict__ gam, const float* __restrict__ bet,
    const float* __restrict__ lw1, const float* __restrict__ lb1,
    const float* __restrict__ lw2, const float* __restrict__ lb2,
    const float* __restrict__ ltw, const float* __restrict__ ltb,
    float* __restrict__ out)
{
  __shared__ float s_nw1[256], s_nw2[256], s_lw1[256], s_lw2[256], s_ltw[256];
  __shared__ float s_nb1[16], s_nb2[16], s_g[16], s_b[16], s_lb1[16], s_lb2[16], s_ltb[16];
  __shared__ float gsum[4], gss[4], mean[4], rstd[4];
  int tid = threadIdx.x;
  if (tid<256){ s_nw1[tid]=nw1[tid]; s_nw2[tid]=nw2[tid]; s_lw1[tid]=lw1[tid];
                s_lw2[tid]=lw2[tid]; s_ltw[tid]=ltw[tid]; }
  if (tid<16){ s_nb1[tid]=nb1[tid]; s_nb2[tid]=nb2[tid]; s_g[tid]=gam[tid]; s_b[tid]=bet[tid];
               s_lb1[tid]=lb1[tid]; s_lb2[tid]=lb2[tid]; s_ltb[tid]=ltb[tid]; }
  if (tid<4){ gsum[tid]=0.f; gss[tid]=0.f; }
  __syncthreads();
  float t2s[4][16];
  float ls[4]={0,0,0,0}, lq[4]={0,0,0,0};
  for (int k=0;k<4;++k){
    int pid = k*1024 + tid;
    float yv[16];
    #pragma unroll
    for (int c=0;c<16;++c) yv[c]=y[pid*16+c];
    float t1[16];
    #pragma unroll
    for (int u=0;u<16;++u){
      float t = s_nb1[u];
      #pragma unroll
      for (int c=0;c<16;++c) t = fmaf(yv[c], s_nw1[u*16+c], t);
      t1[u] = leaky(t);
    }
    #pragma unroll
    for (int c=0;c<16;++c){
      float t = s_nb2[c];
      #pragma unroll
      for (int u=0;u<16;++u) t = fmaf(t1[u], s_nw2[c*16+u], t);
      t = leaky(t);
      t2s[k][c] = t;
      ls[c>>2] += t; lq[c>>2] = fmaf(t,t,lq[c>>2]);
    }
  }
  #pragma unroll
  for (int g=0; g<4; ++g){ atomicAdd(&gsum[g], ls[g]); atomicAdd(&gss[g], lq[g]); }
  __syncthreads();
  if (tid<4){
    float m = gsum[tid] * (1.0f/16384.0f);
    float v = gss[tid] * (1.0f/16384.0f) - m*m;
    mean[tid]=m; rstd[tid]=rsqrtf(v + 1e-5f);
  }
  __syncthreads();
  for (int k=0;k<4;++k){
    int pid = k*1024 + tid;
    float gn[16];
    #pragma unroll
    for (int c=0;c<16;++c){
      int g = c>>2;
      gn[c] = fmaf((t2s[k][c]-mean[g])*rstd[g], s_g[c], s_b[c]);
    }
    float av[16];
    #pragma unroll
    for (int u=0;u<16;++u){
      float t = s_lb1[u];
      #pragma unroll
      for (int c=0;c<16;++c) t = fmaf(gn[c], s_lw1[u*16+c], t);
      av[u] = fmaxf(t, 0.f);
    }
    float fv[16];
    #pragma unroll
    for (int c=0;c<16;++c) fv[c]=feats[pid*16+c];
    #pragma unroll
    for (int o=0;o<16;++o){
      float r = s_lb2[o] + s_ltb[o];
      #pragma unroll
      for (int u=0;u<16;++u) r = fmaf(av[u], s_lw2[o*16+u], r);
      #pragma unroll
      for (int c=0;c<16;++c) r = fmaf(fv[c], s_ltw[o*16+c], r);
      out[pid*16+o] = r;
    }
  }
}

/* -------------------------------- launcher ------------------------------------ */
extern "C" void kernel_launch(void* const* d_in, const int* in_sizes, int n_in,
                              void* d_out, int out_size, void* d_ws, size_t ws_size,
                              hipStream_t stream)
{
  (void)in_sizes; (void)n_in; (void)out_size; (void)ws_size;
  const float* xyz    = (const float*)d_in[0];
  const float* nrm    = (const float*)d_in[1];
  const float* feats  = (const float*)d_in[2];
  const float* os_w1  = (const float*)d_in[3];
  const float* os_b1  = (const float*)d_in[4];
  const float* os_w2  = (const float*)d_in[5];
  const float* os_b2  = (const float*)d_in[6];
  const float* ni_w1  = (const float*)d_in[7];
  const float* ni_b1  = (const float*)d_in[8];
  const float* ni_w2  = (const float*)d_in[9];
  const float* ni_b2  = (const float*)d_in[10];
  const float* gn_in_g= (const float*)d_in[11];
  const float* gn_in_b= (const float*)d_in[12];
  const float* cv_a1  = (const float*)d_in[13];
  const float* cv_b1  = (const float*)d_in[14];
  const float* cv_a2  = (const float*)d_in[15];
  const float* cv_b2  = (const float*)d_in[16];
  const float* no_w1  = (const float*)d_in[17];
  const float* no_b1  = (const float*)d_in[18];
  const float* no_w2  = (const float*)d_in[19];
  const float* no_b2  = (const float*)d_in[20];
  const float* gn_o_g = (const float*)d_in[21];
  const float* gn_o_b = (const float*)d_in[22];
  const float* ll_w1  = (const float*)d_in[23];
  const float* ll_b1  = (const float*)d_in[24];
  const float* ll_w2  = (const float*)d_in[25];
  const float* ll_b2  = (const float*)d_in[26];
  const float* lt_w   = (const float*)d_in[27];
  const float* lt_b   = (const float*)d_in[28];

  float* ws      = (float*)d_ws;
  float* scores  = ws;                     /*  4096 */
  float* nuv     = scores + NPTS;          /* 36864 */
  float* hbuf    = nuv + NPTS*9;           /* 65536 */
  float* ybuf    = hbuf + NPTS*16;         /* 65536 */

  k_scores  <<<NPTS/256, 256, 0, stream>>>(feats, os_w1, os_b1, os_w2, os_b2, scores);
  k_nuv     <<<NPTS/256, 256, 0, stream>>>(xyz, nrm, scores, nuv);
  k_inmlp_gn<<<1, 1024, 0, stream>>>(feats, ni_w1, ni_b1, ni_w2, ni_b2, gn_in_g, gn_in_b, hbuf);
  k_conv    <<<NPTS/16, 256, 0, stream>>>(xyz, nrm, nuv, hbuf, cv_a1, cv_b1, cv_a2, cv_b2, ybuf);
  k_post    <<<1, 1024, 0, stream>>>(ybuf, feats, no_w1, no_b1, no_w2, no_b2,
                                     gn_o_g, gn_o_b, ll_w1, ll_b1, ll_w2, ll_b2,
                                     lt_w, lt_b, (float*)d_out);
}